// CircularAdder_18700287607338
// MI455X (gfx1250) — compile-verified
//
#include <hip/hip_runtime.h>

typedef __attribute__((ext_vector_type(2))) float v2f;
typedef __attribute__((ext_vector_type(8))) float v8f;

#define LOG2E_F 1.44269504088896340736f

// One lane per batch row; each wave owns 32 rows and issues two
// V_WMMA_F32_16X16X4_F32 per time-step (rows 0-15 and 16-31 of the wave).
// v_sin/v_cos take their argument pre-scaled by 1/2pi (revolutions), so the
// angle 2*pi*s/P is fed as s * (1/P) directly.
__global__ __launch_bounds__(256) void CircularAdder_kernel(
    const int* __restrict__ x, const int* __restrict__ y,
    const float* __restrict__ Pp, const float* __restrict__ sharpp,
    const float* __restrict__ lsp, float* __restrict__ out, int B)
{
    constexpr int T  = 64;
    constexpr int RS = T * 10;           // output row stride in floats (640)

    const int gid  = blockIdx.x * blockDim.x + threadIdx.x;   // batch row
    const int lane = threadIdx.x & 31;
    const int r0   = gid & ~31;                               // first row of wave

    const float P     = Pp[0];
    const float sharp = sharpp[0];
    const float ls    = lsp[0];
    const float invP  = 1.0f / P;             // hoisted: one division per thread
    const float ek    = -sharp * LOG2E_F;     // sigmoid: exp2(ek*s + eb)
    const float eb    =  sharp * 9.5f * LOG2E_F;

    // ---- B operand: table^T zero-padded to 4x16 ----
    // VGPR0 = K=0 row (lanes 0-15) / K=2 row (lanes 16-31), VGPR1 = K=1 / K=3.
    const int n = lane & 15;
    const float trev = (float)n * invP;       // revolutions
    const bool bvalid = (lane < 10);          // lanes 10-15 pad N, 16-31 pad K
    v2f Bm;
    Bm.x = bvalid ? __builtin_amdgcn_cosf(trev) : 0.0f;
    Bm.y = bvalid ? __builtin_amdgcn_sinf(trev) : 0.0f;

    float carry = 0.0f;

    // Per-lane output base: rhalf folded in so all 16 store offsets are
    // compile-time immediates (max (7+16)*640*4 = 58880 B < 2^23).
    const int rhalf = (lane >= 16) ? 8 : 0;
    float* pb = out + (long long)(r0 + rhalf) * RS + n;

    const int4* xp = (const int4*)(x + (long long)gid * T);   // 256B-aligned row
    const int4* yp = (const int4*)(y + (long long)gid * T);

    for (int t4 = 0; t4 < T / 4; ++t4) {
        const int4 xv = xp[t4];
        const int4 yv = yp[t4];
        const int xs[4] = {xv.x, xv.y, xv.z, xv.w};
        const int ys[4] = {yv.x, yv.y, yv.z, yv.w};
#pragma unroll
        for (int j = 0; j < 4; ++j) {
            const float s    = (float)(xs[j] + ys[j]) + carry;
            const float revs = s * invP;
            const float cl   = ls * __builtin_amdgcn_cosf(revs);
            const float sl   = ls * __builtin_amdgcn_sinf(revs);

            // A operands (16x4 f32): lanes 0-15 = (K0,K1)=(cos,sin) for M=lane,
            // lanes 16-31 = (K2,K3) zero padding. Upper-half rows reach lanes
            // 0-15 via ds_swizzle (shfl_down by 16).
            v2f A0, A1;
            A0.x = (lane < 16) ? cl : 0.0f;
            A0.y = (lane < 16) ? sl : 0.0f;
            const float cl_hi = __shfl_down(cl, 16, 32);
            const float sl_hi = __shfl_down(sl, 16, 32);
            A1.x = (lane < 16) ? cl_hi : 0.0f;
            A1.y = (lane < 16) ? sl_hi : 0.0f;

            v8f acc0 = {}, acc1 = {};
            acc0 = __builtin_amdgcn_wmma_f32_16x16x4_f32(
                false, A0, false, Bm, (short)0, acc0, false, false);
            acc1 = __builtin_amdgcn_wmma_f32_16x16x4_f32(
                false, A1, false, Bm, (short)0, acc1, false, false);

            // D layout: VGPR v, lanes 0-15 -> M=v, N=lane; lanes 16-31 -> M=v+8.
            if (n < 10) {
#pragma unroll
                for (int v = 0; v < 8; ++v) {
                    pb[v * RS]        = acc0[v];   // rows r0+rhalf+v
                    pb[(v + 16) * RS] = acc1[v];   // rows r0+rhalf+16+v
                }
            }
            pb += 10;                              // advance t (40 B)

            // carry = sigmoid(sharp*(s-9.5)) = rcp(1 + exp2(ek*s + eb))
            carry = __builtin_amdgcn_rcpf(
                1.0f + __builtin_amdgcn_exp2f(__builtin_fmaf(ek, s, eb)));
        }
    }

    // final_carry appended after logits
    out[(long long)B * RS + gid] = carry;
}

extern "C" void kernel_launch(void* const* d_in, const int* in_sizes, int n_in,
                              void* d_out, int out_size, void* d_ws, size_t ws_size,
                              hipStream_t stream) {
    const int*   x     = (const int*)d_in[0];
    const int*   y     = (const int*)d_in[1];
    const float* P     = (const float*)d_in[2];
    const float* sharp = (const float*)d_in[3];
    const float* ls    = (const float*)d_in[4];
    float* out = (float*)d_out;

    const int T = 64;
    const int B = in_sizes[0] / T;   // 65536

    dim3 block(256);
    dim3 grid((B + block.x - 1) / block.x);
    hipLaunchKernelGGL(CircularAdder_kernel, grid, block, 0, stream,
                       x, y, P, sharp, ls, out, B);
}